// RCAutoRec_27393301413999
// MI455X (gfx1250) — compile-verified
//
#include <hip/hip_runtime.h>

// ---------------------------------------------------------------------------
// MI455X / gfx1250, wave32. FP32 WMMA path: V_WMMA_F32_16X16X4_F32.
//
// Layouts (CDNA5 ISA 7.12.2):
//   A (16x4 f32, 2 VGPRs): lane l holds row M=l%16; VGPR0/1 hold K = 2*(l/16)+{0,1}
//   B (4x16 f32, 2 VGPRs): lane l holds col N=l%16; VGPR0/1 hold K = 2*(l/16)+{0,1}
//   C/D (16x16 f32, 8 VGPRs): VGPR r, lane l -> M = r + 8*(l/16), N = l%16
//
// Round-2 changes:
//   * direct 32-bit-offset indexing off kernel-arg pointers (global_load GVS
//     form instead of flat_load -> no DScnt waits in the GEMM loop)
//   * compile-time M/N/K (immediate offsets, no 64-bit addr math, no idiv)
//   * cross-half swap via v_permlanex16 (VALU) instead of ds_bpermute (LDS)
// ---------------------------------------------------------------------------

typedef float v2f __attribute__((ext_vector_type(2)));
typedef float v8f __attribute__((ext_vector_type(8)));

__device__ __forceinline__ float tanh_hw(float x) {
#if __has_builtin(__builtin_amdgcn_tanhf)
  return __builtin_amdgcn_tanhf(x);
#else
  float r;
  // v_tanh_f32 is a TRANS op: one independent op (v_nop) before use/overwrite.
  asm volatile("v_tanh_f32 %0, %1\n\tv_nop" : "=v"(r) : "v"(x));
  return r;
#endif
}

// lane i <-> lane i^16 swap across the two 16-lane halves (wave32), pure VALU.
__device__ __forceinline__ float xor16(float x) {
#if __has_builtin(__builtin_amdgcn_permlanex16)
  int i = __builtin_bit_cast(int, x);
  int r = __builtin_amdgcn_permlanex16(i, i, 0x76543210, 0xfedcba98,
                                       /*fi=*/false, /*bound_ctrl=*/false);
  return __builtin_bit_cast(float, r);
#else
  return __shfl_xor(x, 16, 32);
#endif
}

__device__ __forceinline__ v8f wmma4(v2f a, v2f b, v8f c) {
  // (neg_a, A, neg_b, B, c_mod, C, reuse_a, reuse_b)
  return __builtin_amdgcn_wmma_f32_16x16x4_f32(false, a, false, b, (short)0, c,
                                               false, false);
}

// ---------------------------------------------------------------------------
// C[M,N] = tanh(A[M,K] @ W[K,N] + bias[N]); wave tile = (16*SUB) x 16.
// Grid: ((M/(16*SUB)) * (N/16)) / 8 blocks of 256 threads (8 waves).
// Relies on 192MB L2 for A/W reuse (x, W1, W3, h1, h2 all fit).
// ---------------------------------------------------------------------------
template <int SUB, int M, int N, int K>
__global__ __launch_bounds__(256) void gemm_tanh_k(
    const float* __restrict__ A, const float* __restrict__ W,
    const float* __restrict__ bias, float* __restrict__ C) {
  const int lane = threadIdx.x & 31;
  const int h    = lane >> 4;
  const int l16  = lane & 15;
  const int w    = blockIdx.x * 8 + (threadIdx.x >> 5);
  constexpr int NTN = N >> 4;
  const int mrow = (w / NTN) * (16 * SUB);
  const int ncol = (w % NTN) * 16 + l16;

  // 32-bit element offsets (all < 2^31) -> global_load saddr+voffset form.
  const unsigned abase = (unsigned)(mrow + l16) * K + 2u * h;
  const unsigned bbase = 2u * h * N + (unsigned)ncol;

  v8f acc[SUB];
#pragma unroll
  for (int s = 0; s < SUB; ++s) acc[s] = v8f{};

#pragma unroll 4
  for (int kt = 0; kt < (K >> 2); ++kt) {
    v2f bb;
    bb.x = W[bbase + (unsigned)kt * 4u * N];
    bb.y = W[bbase + (unsigned)kt * 4u * N + N];
#pragma unroll
    for (int s = 0; s < SUB; ++s) {
      v2f aa = *(const v2f*)(A + abase + (unsigned)s * 16u * K + (unsigned)kt * 4u);
      acc[s] = wmma4(aa, bb, acc[s]);
    }
  }

  const float bv = bias[ncol];
#pragma unroll
  for (int s = 0; s < SUB; ++s) {
#pragma unroll
    for (int r = 0; r < 8; ++r) {
      const unsigned row = (unsigned)(mrow + s * 16 + r + 8 * h);
      C[row * N + (unsigned)ncol] = tanh_hw(acc[s][r] + bv);
    }
  }
}

// ---------------------------------------------------------------------------
// out = ENet-blend( H[M,K] @ W3[K,N] + b3 ), fused epilogue.
// ENet e2 (16x16 dense) via WMMA with A = E2w^T (wave-constant):
//   e2^T = E2w^T x e1^T, e1 features live in VGPRs / elements in lanes, so
// B operands come straight from registers + one cross-half permlane swap.
// Wave tile = 64 x 16 -> 40000 waves -> 5000 blocks.
// ---------------------------------------------------------------------------
template <int M, int N, int K>
__global__ __launch_bounds__(256) void gemm3_enet_k(
    const float* __restrict__ H, const float* __restrict__ W3,
    const float* __restrict__ b3, const float* __restrict__ maskp,
    const float* __restrict__ E1w, const float* __restrict__ E1b,
    const float* __restrict__ E2w, const float* __restrict__ E2b,
    const float* __restrict__ E3w, const float* __restrict__ E3b,
    float* __restrict__ out) {
  const int lane = threadIdx.x & 31;
  const int h    = lane >> 4;
  const int l16  = lane & 15;
  const int w    = blockIdx.x * 8 + (threadIdx.x >> 5);
  constexpr int NTN = N >> 4;  // 625
  const int mrow = (w / NTN) * 64;
  const int ncol = (w % NTN) * 16 + l16;

  // ---- GEMM3: 64x16 strip ----
  const unsigned abase = (unsigned)(mrow + l16) * K + 2u * h;
  const unsigned bbase = 2u * h * N + (unsigned)ncol;

  v8f acc[4];
#pragma unroll
  for (int s = 0; s < 4; ++s) acc[s] = v8f{};

#pragma unroll 4
  for (int kt = 0; kt < (K >> 2); ++kt) {  // 128 iters
    v2f bb;
    bb.x = W3[bbase + (unsigned)kt * 4u * N];
    bb.y = W3[bbase + (unsigned)kt * 4u * N + N];
#pragma unroll
    for (int s = 0; s < 4; ++s) {
      v2f aa = *(const v2f*)(H + abase + (unsigned)s * 16u * K + (unsigned)kt * 4u);
      acc[s] = wmma4(aa, bb, acc[s]);
    }
  }

  const float bv = b3[ncol];
#pragma unroll
  for (int s = 0; s < 4; ++s)
#pragma unroll
    for (int r = 0; r < 8; ++r) acc[s][r] += bv;

  // ---- ENet constants ----
  float e1w[16], e1b[16];
#pragma unroll
  for (int k = 0; k < 16; ++k) {  // uniform address -> scalar loads
    e1w[k] = E1w[k];
    e1b[k] = E1b[k];
  }
  // A(t) for wmma t: A[Mj=l16, Kk=4t+2h+{0,1}] = E2w[k][j].
  v2f a2[4];
#pragma unroll
  for (int t = 0; t < 4; ++t) {
    a2[t].x = E2w[(4 * t + 2 * h) * 16 + l16];
    a2[t].y = E2w[(4 * t + 2 * h + 1) * 16 + l16];
  }
  v8f c2;  // C init = E2b[j], j = r + 8h in C layout
#pragma unroll
  for (int r = 0; r < 8; ++r) c2[r] = E2b[r + 8 * h];
  float e3wv[8];
#pragma unroll
  for (int r = 0; r < 8; ++r) e3wv[r] = E3w[r + 8 * h];
  const float e3bs = E3b[0];

  // ---- fused ENet epilogue: 32 elements (one C-VGPR across halves) at a time
#pragma unroll
  for (int s = 0; s < 4; ++s) {
#pragma unroll
    for (int r = 0; r < 8; ++r) {
      const float v = acc[s][r];
      float e1r[16], sw[16];
#pragma unroll
      for (int k = 0; k < 16; ++k)
        e1r[k] = tanh_hw(fmaf(v, e1w[k], e1b[k]));
#pragma unroll
      for (int k = 0; k < 16; ++k) sw[k] = xor16(e1r[k]);

      // Two 16x16x16 gemms: dA covers lane 0-15 elements, dB lanes 16-31.
      v8f dA = c2, dB = c2;
#pragma unroll
      for (int t = 0; t < 4; ++t) {
        v2f bA, bB;
        bA.x = h ? sw[4 * t + 2] : e1r[4 * t];
        bA.y = h ? sw[4 * t + 3] : e1r[4 * t + 1];
        bB.x = h ? e1r[4 * t + 2] : sw[4 * t];
        bB.y = h ? e1r[4 * t + 3] : sw[4 * t + 1];
        dA = wmma4(a2[t], bA, dA);
        dB = wmma4(a2[t], bB, dB);
      }
      // e3: per-lane partial over this half's 8 features, cross-half combine.
      float pA = 0.f, pB = 0.f;
#pragma unroll
      for (int rr = 0; rr < 8; ++rr) {
        pA = fmaf(tanh_hw(dA[rr]), e3wv[rr], pA);
        pB = fmaf(tanh_hw(dB[rr]), e3wv[rr], pB);
      }
      const float yA = pA + xor16(pA);
      const float yB = pB + xor16(pB);
      const float ev = (h ? yB : yA) + e3bs;
      const float y  = ev + 0.8f * v;

      const unsigned row = (unsigned)(mrow + s * 16 + r + 8 * h);
      const unsigned idx = row * N + (unsigned)ncol;
      const float mv = maskp[idx];
      out[idx] = (mv > 0.f) ? y : v;
    }
  }
}

extern "C" void kernel_launch(void* const* d_in, const int* in_sizes, int n_in,
                              void* d_out, int out_size, void* d_ws,
                              size_t ws_size, hipStream_t stream) {
  const float* x    = (const float*)d_in[0];
  const float* mask = (const float*)d_in[1];
  const float* W1   = (const float*)d_in[2];
  const float* b1   = (const float*)d_in[3];
  const float* W2   = (const float*)d_in[4];
  const float* b2   = (const float*)d_in[5];
  const float* W3   = (const float*)d_in[6];
  const float* b3   = (const float*)d_in[7];
  const float* E1w  = (const float*)d_in[8];
  const float* E1b  = (const float*)d_in[9];
  const float* E2w  = (const float*)d_in[10];
  const float* E2b  = (const float*)d_in[11];
  const float* E3w  = (const float*)d_in[12];
  const float* E3b  = (const float*)d_in[13];
  float* out = (float*)d_out;

  constexpr int B = 4096, D = 10000, Hd = 512;
  float* h1 = (float*)d_ws;               // 8 MB
  float* h2 = h1 + (size_t)B * Hd;        // 8 MB

  // GEMM1: 4096x512, K=10000. SUB=2 -> (4096/32)*(512/16)=4096 waves -> 512 blk
  gemm_tanh_k<2, B, Hd, D><<<512, 256, 0, stream>>>(x, W1, b1, h1);
  // GEMM2: 4096x512, K=512.
  gemm_tanh_k<2, B, Hd, Hd><<<512, 256, 0, stream>>>(h1, W2, b2, h2);
  // GEMM3 + fused ENet: (4096/64)*(10000/16)=40000 waves -> 5000 blocks.
  gemm3_enet_k<B, D, Hd><<<5000, 256, 0, stream>>>(h2, W3, b3, mask, E1w, E1b,
                                                   E2w, E2b, E3w, E3b, out);
}